// DiceLoss_10222022164824
// MI455X (gfx1250) — compile-verified
//
#include <hip/hip_runtime.h>

// ---------------------------------------------------------------------------
// Dice loss, MI455X (gfx1250, wave32).
// Phase 1: streaming partial sums (HBM-bound, float4/int4 b128 loads),
//          block tail reduced with V_WMMA_F32_16X16X4_F32 (B = ones trick).
// Phase 2: single-wave deterministic fold of block partials (WMMA again)
//          + dice epilogue.
// ---------------------------------------------------------------------------

typedef float v2f __attribute__((ext_vector_type(2)));
typedef float v8f __attribute__((ext_vector_type(8)));

#define NUM_CLASS 33
#define NSPAT     (96 * 96 * 96)          // 884736 voxels per batch
#define CHUNKS    32                      // chunks per (b, c)
#define CH        (NSPAT / CHUNKS)        // 27648 elements per block
#define TPB       256                     // 8 waves of 32
#define ITERS     (CH / (TPB * 4))        // 27 float4 iterations per thread
#define NPART     (2 * CHUNKS)            // 64 block-partials per class
#define SMOOTH    1e-5f

// Sum 64 f32 values held as {a0, a1} per lane across a full wave32.
// A layout (ISA 7.12.2, 16x4 f32): lane l<16 -> A[l,0]=a0, A[l,1]=a1;
// lane l>=16 -> A[l-16,2]=a0, A[l-16,3]=a1.  With B == all-ones (layout-
// invariant), D[m,n] = row-sum s_m replicated over n.  D layout: VGPR r
// holds s_r (lanes 0-15) and s_{r+8} (lanes 16-31), so adding the 8 D
// registers and one shfl_xor(16) yields the full 64-element sum in every
// lane, all in f32.  Requires EXEC all-1s (callers guarantee a full wave).
__device__ __forceinline__ float wmma_sum64(float a0, float a1) {
    v2f a; a.x = a0; a.y = a1;
    v2f b; b.x = 1.0f; b.y = 1.0f;     // all-ones B: layout-independent
    v8f c = {};
    v8f d = __builtin_amdgcn_wmma_f32_16x16x4_f32(
        /*neg_a=*/false, a, /*neg_b=*/false, b,
        /*c_mod=*/(short)0, c, /*reuse_a=*/false, /*reuse_b=*/false);
    float t = ((d[0] + d[1]) + (d[2] + d[3])) + ((d[4] + d[5]) + (d[6] + d[7]));
    t += __shfl_xor(t, 16, 32);        // combine s_0..7 with s_8..15
    return t;                          // broadcast to all 32 lanes
}

// Phase 1: grid = (CHUNKS, NUM_CLASS, 2).  Each block streams one 27648-
// element chunk of one class and emits 3 exact partials (sq, intersect,
// count) to part[q][c][b*CHUNKS+chunk].
__global__ void __launch_bounds__(TPB)
dice_partial_kernel(const float* __restrict__ outputs,
                    const int*   __restrict__ label,
                    float*       __restrict__ part) {
    const int chunk = blockIdx.x;
    const int c     = blockIdx.y;
    const int b     = blockIdx.z;
    const int tid   = threadIdx.x;

    const float4* __restrict__ op = (const float4*)(
        outputs + (size_t)(b * NUM_CLASS + c) * NSPAT + (size_t)chunk * CH);
    const int4* __restrict__ lp = (const int4*)(
        label + (size_t)b * NSPAT + (size_t)chunk * CH);

    float acc_sq = 0.0f, acc_in = 0.0f, acc_cn = 0.0f;

#pragma unroll 3
    for (int it = 0; it < ITERS; ++it) {
        const int idx = it * TPB + tid;
        const float4 x  = op[idx];      // global_load_b128, fully coalesced
        const int4   la = lp[idx];      // labels: L2-resident after 1st class

        const float m0 = (la.x == c) ? 1.0f : 0.0f;
        const float m1 = (la.y == c) ? 1.0f : 0.0f;
        const float m2 = (la.z == c) ? 1.0f : 0.0f;
        const float m3 = (la.w == c) ? 1.0f : 0.0f;

        acc_sq = fmaf(x.x, x.x, acc_sq);
        acc_sq = fmaf(x.y, x.y, acc_sq);
        acc_sq = fmaf(x.z, x.z, acc_sq);
        acc_sq = fmaf(x.w, x.w, acc_sq);

        acc_in = fmaf(x.x, m0, acc_in);
        acc_in = fmaf(x.y, m1, acc_in);
        acc_in = fmaf(x.z, m2, acc_in);
        acc_in = fmaf(x.w, m3, acc_in);

        acc_cn += (m0 + m1) + (m2 + m3);
    }

    __shared__ float red[3][TPB];
    red[0][tid] = acc_sq;
    red[1][tid] = acc_in;
    red[2][tid] = acc_cn;
    __syncthreads();

    if (tid < 32) {                    // wave 0 only: EXEC all-1s for WMMA
        const int j = b * CHUNKS + chunk;
#pragma unroll
        for (int q = 0; q < 3; ++q) {
            const float a0 = ((red[q][tid]       + red[q][tid + 64]) +
                              (red[q][tid + 128] + red[q][tid + 192]));
            const float a1 = ((red[q][tid + 32]  + red[q][tid + 96]) +
                              (red[q][tid + 160] + red[q][tid + 224]));
            const float s = wmma_sum64(a0, a1);   // 256 -> 1, exact f32 order
            if (tid == 0)
                part[(q * NUM_CLASS + c) * NPART + j] = s;
        }
    }
}

// Phase 2: one wave.  Folds the 64 block partials per (q, c) with WMMA,
// then the dice epilogue.  Fully deterministic (no atomics anywhere).
__global__ void __launch_bounds__(32)
dice_final_kernel(const float* __restrict__ part,
                  float*       __restrict__ out) {
    const int lane = threadIdx.x;      // 32 threads == 1 wave, EXEC all-1s
    float acc = 0.0f;
    for (int c = 0; c < NUM_CLASS; ++c) {
        float s[3];
#pragma unroll
        for (int q = 0; q < 3; ++q) {
            const float* p = part + (q * NUM_CLASS + c) * NPART;
            s[q] = wmma_sum64(p[lane], p[lane + 32]);   // 64 -> 1
        }
        const float dice = (2.0f * s[1] + SMOOTH) / (s[0] + s[2] + SMOOTH);
        acc += 1.0f - dice;            // identical in every lane
    }
    if (lane == 0)
        out[0] = acc * (1.0f / (float)NUM_CLASS);
}

extern "C" void kernel_launch(void* const* d_in, const int* in_sizes, int n_in,
                              void* d_out, int out_size, void* d_ws, size_t ws_size,
                              hipStream_t stream) {
    (void)in_sizes; (void)n_in; (void)out_size; (void)ws_size;

    const float* outputs = (const float*)d_in[0];
    const int*   label   = (const int*)d_in[1];
    float*       part    = (float*)d_ws;          // 3*33*64 floats = 25 KB
    float*       out     = (float*)d_out;

    dim3 grid(CHUNKS, NUM_CLASS, 2);              // 2112 blocks
    dice_partial_kernel<<<grid, TPB, 0, stream>>>(outputs, label, part);
    dice_final_kernel<<<1, 32, 0, stream>>>(part, out);
}